// act_quantizer_28647431865075
// MI455X (gfx1250) — compile-verified
//
#include <hip/hip_runtime.h>
#include <math.h>
#include <stdint.h>

// ---------------------------------------------------------------------------
// act_quantizer (phase 3): x[1,8192,4096] f32 -> (mxfp4, asym_int8, sym_int8)
// Pure streaming op: 512 MB total traffic -> ~22us at 23.3 TB/s. Memory-bound
// by >100x, so the kernel is built around CDNA5 data movement:
//   * global_load_async_to_lds_b128 (ASYNCcnt) double-buffered input stream
//   * ds_swizzle wave32 xor-reductions for the 128-elem and 32-elem groups
//   * non-temporal global_store_b128 for the 384 MB of write-once output
// No matmul in this op -> WMMA not applicable.
// ---------------------------------------------------------------------------

typedef float v4f __attribute__((ext_vector_type(4)));

#define WAVES_PER_BLOCK 8
#define NG 4          // 128-element groups per wave (pipeline depth NG, 2 LDS bufs)
#define GROUP 128     // int-quant group size = 32 lanes * 4 floats

// ---- wave32 xor-shuffle via ds_swizzle (group-of-32 mode, and=0x1f, xor=XM)
template <int XM>
__device__ __forceinline__ float lane_xor(float x) {
  return __int_as_float(
      __builtin_amdgcn_ds_swizzle(__float_as_int(x), 0x1f | (XM << 10)));
}

__device__ __forceinline__ float wave_max32(float x) {
  x = fmaxf(x, lane_xor<16>(x));
  x = fmaxf(x, lane_xor<8>(x));
  x = fmaxf(x, lane_xor<4>(x));
  x = fmaxf(x, lane_xor<2>(x));
  x = fmaxf(x, lane_xor<1>(x));
  return x;
}
__device__ __forceinline__ float wave_min32(float x) {
  x = fminf(x, lane_xor<16>(x));
  x = fminf(x, lane_xor<8>(x));
  x = fminf(x, lane_xor<4>(x));
  x = fminf(x, lane_xor<2>(x));
  x = fminf(x, lane_xor<1>(x));
  return x;
}
// max over 8 consecutive lanes (one MXFP4 group of 32 elements)
__device__ __forceinline__ float grp8_max(float x) {
  x = fmaxf(x, lane_xor<4>(x));
  x = fmaxf(x, lane_xor<2>(x));
  x = fmaxf(x, lane_xor<1>(x));
  return x;
}

// ---- MXFP4 E2M1 encode of one non-negative value. norm = |v| / X, X = 2^se.
// Mirrors the reference math; floor(log2) done via exponent bits (exact for
// normals; zero/denormal norm falls into the subnormal branch as intended).
__device__ __forceinline__ float mx_enc(float norm, float X) {
  int e = (int)((__float_as_uint(norm) >> 23) & 0xff) - 127;
  bool is_sub = (e < 0);
  int ec = (e < 0) ? 0 : ((e > 2) ? 2 : e);
  float m = rintf(ldexpf(norm, 1 - ec) - 2.0f);  // round((norm/2^e - 1)*2), RNE
  bool carry = (m == 2.0f);
  bool satu = carry && (ec == 2);
  bool ni = carry && !satu;
  int e2 = ni ? (ec + 1) : ec;
  m = ni ? 0.0f : m;
  m = satu ? 1.0f : m;
  m = fminf(fmaxf(m, 0.0f), 1.0f) * 0.5f + 1.0f;
  // subnormal branch: round(norm*2) with carry folding to 1.0
  float ms = rintf(norm * 2.0f);
  ms = (ms == 2.0f) ? 1.0f : (fminf(fmaxf(ms, 0.0f), 1.0f) * 0.5f);
  float mm = is_sub ? ms : m;
  int ee = is_sub ? 0 : e2;
  return ldexpf(mm * X, ee);  // 2^e * m * X
}

__global__ __launch_bounds__(256) void act_quant3_kernel(
    const float* __restrict__ in, float* __restrict__ out_mx,
    float* __restrict__ out_as, float* __restrict__ out_sy,
    long long ngroups) {
  __shared__ __align__(16) float smem[WAVES_PER_BLOCK * 2 * GROUP];  // 8 KB

  const int lane = threadIdx.x & 31;
  const int wave = threadIdx.x >> 5;
  const long long wgid = (long long)blockIdx.x * WAVES_PER_BLOCK + wave;
  const long long g0 = wgid * NG;
  if (g0 >= ngroups) return;
  const long long rem = ngroups - g0;
  const int ngmine = (rem < (long long)NG) ? (int)rem : NG;

  // Wave-relative LDS byte address of this wave's two 512B buffers.
  // (Flat pointer to LDS has the LDS offset in its low 32 bits.)
  const unsigned lds0 = (unsigned)(size_t)(&smem[wave * 2 * GROUP]);
  const void* gbase = (const void*)in;

  // ---- prologue: async-load group g0 into buffer 0 (ASYNCcnt path)
  {
    unsigned ldsa = lds0 + (unsigned)(lane * 16);
    unsigned voff = (unsigned)((g0 * GROUP + lane * 4) * sizeof(float));
    asm volatile("global_load_async_to_lds_b128 %0, %1, %2"
                 :
                 : "v"(ldsa), "v"(voff), "s"(gbase)
                 : "memory");
  }

  for (int i = 0; i < ngmine; ++i) {
    const long long gi = g0 + i;

    // ---- prefetch next group into the other buffer, then wait for current
    if (i + 1 < ngmine) {
      unsigned ldsa =
          lds0 + (unsigned)(((i + 1) & 1) * (GROUP * 4)) + (unsigned)(lane * 16);
      unsigned voff = (unsigned)(((gi + 1) * GROUP + lane * 4) * sizeof(float));
      asm volatile("global_load_async_to_lds_b128 %0, %1, %2"
                   :
                   : "v"(ldsa), "v"(voff), "s"(gbase)
                   : "memory");
      asm volatile("s_wait_asynccnt 0x1" ::: "memory");  // current buf landed
    } else {
      asm volatile("s_wait_asynccnt 0x0" ::: "memory");
    }

    // ---- read this lane's 4 elements from LDS (ds_load_b128)
    const v4f v =
        *(const v4f*)&smem[wave * 2 * GROUP + (i & 1) * GROUP + lane * 4];
    const float a0 = fabsf(v.x), a1 = fabsf(v.y), a2 = fabsf(v.z),
                a3 = fabsf(v.w);
    const float lmaxa = fmaxf(fmaxf(a0, a1), fmaxf(a2, a3));
    const float lmax = fmaxf(fmaxf(v.x, v.y), fmaxf(v.z, v.w));
    const float lmin = fminf(fminf(v.x, v.y), fminf(v.z, v.w));

    // ================= symmetric int8, group of 128 (whole wave) ===========
    const float gmaxa = wave_max32(lmaxa);
    const float s_scale = gmaxa * (1.0f / 127.0f);
    const float s_inv = 127.0f / gmaxa;
    v4f sy;
    sy.x = copysignf(fminf(rintf(a0 * s_inv), 127.0f) * s_scale, v.x);
    sy.y = copysignf(fminf(rintf(a1 * s_inv), 127.0f) * s_scale, v.y);
    sy.z = copysignf(fminf(rintf(a2 * s_inv), 127.0f) * s_scale, v.z);
    sy.w = copysignf(fminf(rintf(a3 * s_inv), 127.0f) * s_scale, v.w);

    // ================= asymmetric int8, group of 128 =======================
    float dmax = wave_max32(lmax);
    float dmin = wave_min32(lmin);
    const bool bz = (dmax == 0.0f) && (dmin == 0.0f);
    dmax = bz ? 1.0f : dmax;
    dmin = bz ? -1.0f : dmin;
    const float a_scale = (dmax - dmin) * (1.0f / 255.0f);
    const float a_inv = 1.0f / a_scale;
    const float zp = rintf(-dmin * a_inv);
    v4f as;
    as.x = (fminf(fmaxf(rintf(v.x * a_inv) + zp, 0.0f), 255.0f) - zp) * a_scale;
    as.y = (fminf(fmaxf(rintf(v.y * a_inv) + zp, 0.0f), 255.0f) - zp) * a_scale;
    as.z = (fminf(fmaxf(rintf(v.z * a_inv) + zp, 0.0f), 255.0f) - zp) * a_scale;
    as.w = (fminf(fmaxf(rintf(v.w * a_inv) + zp, 0.0f), 255.0f) - zp) * a_scale;

    // ================= MXFP4, group of 32 (8 lanes) ========================
    const float grmax = grp8_max(lmaxa) + 1.1920929e-7f;  // + f32 eps
    // shared_e = floor(log2(grmax)) - 2, exact via exponent bits
    const int se = (int)((__float_as_uint(grmax) >> 23) & 0xff) - 129;
    const float X = ldexpf(1.0f, se);
    const float invX = ldexpf(1.0f, -se);
    v4f mx;
    mx.x = copysignf(mx_enc(a0 * invX, X), v.x);
    mx.y = copysignf(mx_enc(a1 * invX, X), v.y);
    mx.z = copysignf(mx_enc(a2 * invX, X), v.z);
    mx.w = copysignf(mx_enc(a3 * invX, X), v.w);

    // ---- non-temporal b128 stores (write-once 384 MB; keep out of L2)
    const size_t off = (size_t)gi * GROUP + (size_t)(lane * 4);
    __builtin_nontemporal_store(mx, (v4f*)(out_mx + off));
    __builtin_nontemporal_store(as, (v4f*)(out_as + off));
    __builtin_nontemporal_store(sy, (v4f*)(out_sy + off));
  }
}

extern "C" void kernel_launch(void* const* d_in, const int* in_sizes, int n_in,
                              void* d_out, int out_size, void* d_ws,
                              size_t ws_size, hipStream_t stream) {
  (void)n_in;
  (void)out_size;
  (void)d_ws;
  (void)ws_size;
  const float* x = (const float*)d_in[0];
  const long long n = (long long)in_sizes[0];  // 8192*4096 = 33,554,432
  float* out = (float*)d_out;
  float* out_mx = out;           // output order: mxfp4, asym, sym
  float* out_as = out + n;
  float* out_sy = out + 2 * n;

  const long long ngroups = n / GROUP;  // 262,144
  const long long waves = (ngroups + NG - 1) / NG;
  const int blocks = (int)((waves + WAVES_PER_BLOCK - 1) / WAVES_PER_BLOCK);

  act_quant3_kernel<<<blocks, 32 * WAVES_PER_BLOCK, 0, stream>>>(
      x, out_mx, out_as, out_sy, ngroups);
}